// LightGCN_60644938219889
// MI455X (gfx1250) — compile-verified
//
#include <hip/hip_runtime.h>

// ---------------------------------------------------------------------------
// LightGCN propagation on MI455X (gfx1250).
// Memory/latency bound: emb table (51.2MB) is L2-resident; work is random
// 256B row gathers + f32 scatter-atomics. Edge lists are double-buffer
// staged to LDS via the CDNA5 async-to-LDS path (ASYNCcnt), overlapping the
// next chunk's HBM stream with the current chunk's gather/atomic processing.
// Disasm-verified: global_load_async_to_lds_b128 + s_wait_asynccnt + non-
// returning f32 atomics (STOREcnt-tracked => global_atomic_add_f32).
// ---------------------------------------------------------------------------

#define DDIM   64          // embedding dim (ref: D = 64), row = 256B, aligned
#define CHUNK  1024        // edges staged per block per step
#define TPB    256         // 8 wave32 per block
#define CPB    2           // chunks per block (grid-stride depth)

#if defined(__HIP_DEVICE_COMPILE__) && defined(__has_builtin)
#  if __has_builtin(__builtin_amdgcn_global_load_async_to_lds_b128) && \
      __has_builtin(__builtin_amdgcn_s_wait_asynccnt)
#    define HAVE_ASYNC_LDS 1
#  endif
#endif

#ifdef HAVE_ASYNC_LDS
typedef int v4i __attribute__((vector_size(4 * sizeof(int))));
__device__ __forceinline__ void async_ld_b128(void* g, void* l) {
  __builtin_amdgcn_global_load_async_to_lds_b128(
      (__attribute__((address_space(1))) v4i*)g,
      (__attribute__((address_space(3))) v4i*)l,
      /*offset=*/0, /*cpol=*/0);
}
#endif

// One edge-slice step: 16 lanes per edge, lane l covers floats [4l, 4l+3].
__device__ __forceinline__ void edge_step(const float* __restrict__ src,
                                          float* __restrict__ dst,
                                          int c, int r, float v, int l) {
  const float4 s4 = *(const float4*)(src + (size_t)c * DDIM + (size_t)l * 4);
  float* dp = dst + (size_t)r * DDIM + (size_t)l * 4;
  unsafeAtomicAdd(dp + 0, v * s4.x);   // -> global_atomic_add_f32
  unsafeAtomicAdd(dp + 1, v * s4.y);
  unsafeAtomicAdd(dp + 2, v * s4.z);
  unsafeAtomicAdd(dp + 3, v * s4.w);
}

// COO SpMM: dst[row[e]] += vals[e] * src[col[e]]  (dst pre-zeroed)
__global__ void __launch_bounds__(TPB)
k_spmm_stage(const float* __restrict__ src, float* __restrict__ dst,
             const float* __restrict__ vals, const int* __restrict__ row,
             const int* __restrict__ col, int E, int nChunks) {
  __shared__ __align__(16) int   s_row[2][CHUNK];
  __shared__ __align__(16) int   s_col[2][CHUNK];
  __shared__ __align__(16) float s_val[2][CHUNK];

  const int t      = threadIdx.x;      // 0..255
  const int eoff   = t * 4;            // 16B slice per thread per array
  const int grp    = t >> 4;           // 16 edge-groups per block pass
  const int l      = t & 15;           // lane-in-group
  const int stride = gridDim.x;

#ifdef HAVE_ASYNC_LDS
  int ch  = blockIdx.x;
  int cur = 0;
  { // prologue: stage first chunk into buffer 0
    int g = ch * CHUNK + eoff;
    if (g > E - 4) g = E - 4;          // clamp; E is a multiple of 4, and any
                                       // mis-mapped tail slot has edge id >= E
                                       // which only the guarded tail loop sees.
    async_ld_b128((void*)(row  + g), &s_row[0][eoff]);
    async_ld_b128((void*)(col  + g), &s_col[0][eoff]);
    async_ld_b128((void*)(vals + g), &s_val[0][eoff]);
  }
  for (; ch < nChunks; ch += stride) {
    const int nxt = ch + stride;
    if (nxt < nChunks) {               // stage next chunk into other buffer
      int g = nxt * CHUNK + eoff;
      if (g > E - 4) g = E - 4;
      async_ld_b128((void*)(row  + g), &s_row[cur ^ 1][eoff]);
      async_ld_b128((void*)(col  + g), &s_col[cur ^ 1][eoff]);
      async_ld_b128((void*)(vals + g), &s_val[cur ^ 1][eoff]);
      // async loads complete in order: <=3 outstanding => current chunk done
      __builtin_amdgcn_s_wait_asynccnt(3);
    } else {
      __builtin_amdgcn_s_wait_asynccnt(0);
    }
    __syncthreads();                   // staged data visible block-wide
    const int base = ch * CHUNK;
    if (base + CHUNK <= E) {           // fast path: full chunk, no bounds check
#pragma unroll 8
      for (int i = grp; i < CHUNK; i += 16)
        edge_step(src, dst, s_col[cur][i], s_row[cur][i], s_val[cur][i], l);
    } else {                           // tail chunk
      for (int i = grp; i < CHUNK; i += 16) {
        if (base + i >= E) break;
        edge_step(src, dst, s_col[cur][i], s_row[cur][i], s_val[cur][i], l);
      }
    }
    __syncthreads();                   // all readers done before buffer reuse
    cur ^= 1;
  }
#else
  for (int ch = blockIdx.x; ch < nChunks; ch += stride) {
    int g = ch * CHUNK + eoff;
    if (g > E - 4) g = E - 4;
    *(int4*)  &s_row[0][eoff] = *(const int4*)  (row  + g);
    *(int4*)  &s_col[0][eoff] = *(const int4*)  (col  + g);
    *(float4*)&s_val[0][eoff] = *(const float4*)(vals + g);
    __syncthreads();
    const int base = ch * CHUNK;
    if (base + CHUNK <= E) {
#pragma unroll 8
      for (int i = grp; i < CHUNK; i += 16)
        edge_step(src, dst, s_col[0][i], s_row[0][i], s_val[0][i], l);
    } else {
      for (int i = grp; i < CHUNK; i += 16) {
        if (base + i >= E) break;
        edge_step(src, dst, s_col[0][i], s_row[0][i], s_val[0][i], l);
      }
    }
    __syncthreads();
  }
#endif
}

// acc = bufA = concat(user_emb, item_emb); bufB = 0
__global__ void k_init(const float* __restrict__ ue, const float* __restrict__ ie,
                       float* __restrict__ acc, float* __restrict__ bufA,
                       float* __restrict__ bufB, int nUserFloats, int nTotFloats) {
  int i = (blockIdx.x * blockDim.x + threadIdx.x) * 4;
  if (i >= nTotFloats) return;
  float4 v = (i < nUserFloats) ? *(const float4*)(ue + i)
                               : *(const float4*)(ie + (i - nUserFloats));
  *(float4*)(acc + i)  = v;
  *(float4*)(bufA + i) = v;
  *(float4*)(bufB + i) = make_float4(0.f, 0.f, 0.f, 0.f);
}

// acc = (acc + addv) * scale ; zbuf = 0  (prepares next layer's scatter dst)
__global__ void k_addzero(float* __restrict__ acc, const float* __restrict__ addv,
                          float* __restrict__ zbuf, float scale, int nTotFloats) {
  int i = (blockIdx.x * blockDim.x + threadIdx.x) * 4;
  if (i >= nTotFloats) return;
  float4 a = *(const float4*)(acc + i);
  float4 b = *(const float4*)(addv + i);
  a.x = (a.x + b.x) * scale;
  a.y = (a.y + b.y) * scale;
  a.z = (a.z + b.z) * scale;
  a.w = (a.w + b.w) * scale;
  *(float4*)(acc + i)  = a;
  *(float4*)(zbuf + i) = make_float4(0.f, 0.f, 0.f, 0.f);
}

extern "C" void kernel_launch(void* const* d_in, const int* in_sizes, int n_in,
                              void* d_out, int out_size, void* d_ws, size_t ws_size,
                              hipStream_t stream) {
  const float* ue   = (const float*)d_in[0];   // user_emb  [N_USERS*64]
  const float* ie   = (const float*)d_in[1];   // item_emb  [N_ITEMS*64]
  const float* vals = (const float*)d_in[2];   // vals      [E]
  const int*   row  = (const int*)  d_in[3];   // row       [E]
  const int*   col  = (const int*)  d_in[4];   // col       [E]

  const int nUserFloats = in_sizes[0];
  const int nTotFloats  = in_sizes[0] + in_sizes[1];   // 12.8M floats
  const int E           = in_sizes[2];                 // 1.2M edges

  float* acc  = (float*)d_out;
  float* bufA = (float*)d_ws;
  float* bufB = bufA + nTotFloats;                     // 2 x 51.2MB in d_ws

  const int nVec    = nTotFloats / 4;
  const int vBlks   = (nVec + TPB - 1) / TPB;
  const int nChunks = (E + CHUNK - 1) / CHUNK;
  const int eBlks   = (nChunks + CPB - 1) / CPB;       // CPB chunks per block

  k_init<<<vBlks, TPB, 0, stream>>>(ue, ie, acc, bufA, bufB,
                                    nUserFloats, nTotFloats);

  for (int layer = 0; layer < 3; ++layer) {
    const float* s  = (layer & 1) ? bufB : bufA;   // gather source
    float*       d  = (layer & 1) ? bufA : bufB;   // scatter dest (pre-zeroed)
    float*       z  = (layer & 1) ? bufB : bufA;   // zero for next layer
    k_spmm_stage<<<eBlks, TPB, 0, stream>>>(s, d, vals, row, col, E, nChunks);
    const float scale = (layer == 2) ? 0.25f : 1.0f;  // fold final mean
    k_addzero<<<vBlks, TPB, 0, stream>>>(acc, d, z, scale, nTotFloats);
  }
}